// PadeModel_10909216932199
// MI455X (gfx1250) — compile-verified
//
#include <hip/hip_runtime.h>

// MI455X / gfx1250, wave32. FP32 elementwise pipeline + V_WMMA_F32_16X16X4_F32
// for the 8->4 complex dense (real-expanded [16x16]x[16x8] GEMM per 16 samples).

typedef __attribute__((ext_vector_type(2))) float v2f;
typedef __attribute__((ext_vector_type(8))) float v8f;

struct cplx { float re, im; };

__device__ __forceinline__ cplx cfma(cplx a, cplx b, cplx c) {
  cplx r;
  r.re = __builtin_fmaf(a.re, b.re, __builtin_fmaf(-a.im, b.im, c.re));
  r.im = __builtin_fmaf(a.re, b.im, __builtin_fmaf( a.im, b.re, c.im));
  return r;
}

__device__ __forceinline__ cplx cdiv(cplx n, cplx d) {
  float inv = __builtin_amdgcn_rcpf(__builtin_fmaf(d.re, d.re, d.im * d.im));
  cplx r;
  r.re = __builtin_fmaf(n.re, d.re,  n.im * d.im) * inv;
  r.im = __builtin_fmaf(n.im, d.re, -n.re * d.im) * inv;
  return r;
}

__device__ __forceinline__ cplx pade(cplx h, const cplx a[5], const cplx q[4]) {
  cplx num = a[4];
  num = cfma(num, h, a[3]); num = cfma(num, h, a[2]);
  num = cfma(num, h, a[1]); num = cfma(num, h, a[0]);
  cplx den = q[3];
  den = cfma(den, h, q[2]); den = cfma(den, h, q[1]); den = cfma(den, h, q[0]);
  cplx one = {1.0f, 0.0f};
  den = cfma(den, h, one);
  return cdiv(num, den);
}

// Real expansion of the complex 8->4 weight matrix: in-feature k (k<8: Re h_k,
// k>=8: Im h_{k-8}), out-feature n (n<4: Re y_n, 4<=n<8: Im y_{n-4}, n>=8: 0).
__device__ __forceinline__ float wreal(int k, int n,
                                       const float* __restrict__ W2r,
                                       const float* __restrict__ W2i) {
  if (n >= 8) return 0.0f;
  int j = n & 3;
  int i = (k < 8) ? k : (k - 8);
  float wr = W2r[i * 4 + j];
  float wi = W2i[i * 4 + j];
  if (n < 4) return (k < 8) ? wr : -wi;   // Re out
  return (k < 8) ? wi : wr;               // Im out
}

#define ROWQ 20   // dwords per LDS row (80 B: 16B-aligned, bank-spread)

__global__ __launch_bounds__(256) void pade_net_kernel(
    const float* __restrict__ x,
    const float* __restrict__ W1r, const float* __restrict__ W1i,
    const float* __restrict__ b1r, const float* __restrict__ b1i,
    const float* __restrict__ W2r, const float* __restrict__ W2i,
    const float* __restrict__ b2r, const float* __restrict__ b2i,
    const float* __restrict__ W3r, const float* __restrict__ W3i,
    const float* __restrict__ b3r, const float* __restrict__ b3i,
    const float* __restrict__ a1r, const float* __restrict__ a1i,
    const float* __restrict__ q1r, const float* __restrict__ q1i,
    const float* __restrict__ a2r, const float* __restrict__ a2i,
    const float* __restrict__ q2r, const float* __restrict__ q2i,
    float* __restrict__ out, int n)
{
  __shared__ float lds[8 * 32 * ROWQ];   // 8 waves x 32 rows x 20 dwords = 20 KB

  const int  tid  = threadIdx.x;
  const int  lane = tid & 31;
  const int  ln   = lane & 15;
  const bool low  = lane < 16;
  float* wlds = &lds[(tid >> 5) * 32 * ROWQ];

  // ---- uniform weights / Pade coefficients -> (scalar) registers ----
  cplx w1[8], bb1[8];
  #pragma unroll
  for (int j = 0; j < 8; ++j) { w1[j] = {W1r[j], W1i[j]}; bb1[j] = {b1r[j], b1i[j]}; }
  cplx a1[5], a2[5];
  #pragma unroll
  for (int j = 0; j < 5; ++j) { a1[j] = {a1r[j], a1i[j]}; a2[j] = {a2r[j], a2i[j]}; }
  cplx q1[4], q2[4];
  #pragma unroll
  for (int j = 0; j < 4; ++j) { q1[j] = {q1r[j], q1i[j]}; q2[j] = {q2r[j], q2i[j]}; }
  cplx w3[4];
  #pragma unroll
  for (int j = 0; j < 4; ++j) w3[j] = {W3r[j], W3i[j]};
  cplx bb3 = {b3r[0], b3i[0]};

  // ---- per-lane B-matrix (layer-2 weights, K=in-feat rows striped across halves) ----
  v2f Bm[4];
  #pragma unroll
  for (int t = 0; t < 4; ++t) {
    int kx = 4 * t + (low ? 0 : 2);
    Bm[t].x = wreal(kx,     ln, W2r, W2i);
    Bm[t].y = wreal(kx + 1, ln, W2r, W2i);
  }
  float biasN = (ln < 4) ? b2r[ln] : ((ln < 8) ? b2i[ln - 4] : 0.0f);
  v8f cinit;
  #pragma unroll
  for (int r = 0; r < 8; ++r) cinit[r] = biasN;

  const int stride = gridDim.x * blockDim.x;
  for (int s = blockIdx.x * blockDim.x + tid; s < n; s += stride) {
    // fugacity z = exp(mu), mu = xr + i*xi
    float2 xv = reinterpret_cast<const float2*>(x)[s];
    float er = __expf(xv.x);
    float sn, cs; __sincosf(xv.y, &sn, &cs);
    cplx z = {er * cs, er * sn};

    // layer 1 (1->8 complex) + Pade
    cplx h[8];
    #pragma unroll
    for (int j = 0; j < 8; ++j) h[j] = pade(cfma(z, w1[j], bb1[j]), a1, q1);

    // ---- forward marshal: row = lane, feats [Re0..7, Im0..7] ----
    float4* rowp = reinterpret_cast<float4*>(&wlds[lane * ROWQ]);
    rowp[0] = make_float4(h[0].re, h[1].re, h[2].re, h[3].re);
    rowp[1] = make_float4(h[4].re, h[5].re, h[6].re, h[7].re);
    rowp[2] = make_float4(h[0].im, h[1].im, h[2].im, h[3].im);
    rowp[3] = make_float4(h[4].im, h[5].im, h[6].im, h[7].im);

    // ---- layer 2 via 8x v_wmma_f32_16x16x4_f32 (two M=16 sample groups) ----
    v8f acc0 = cinit, acc1 = cinit;
    #pragma unroll
    for (int t = 0; t < 4; ++t) {
      int kof = 4 * t + (low ? 0 : 2);   // A layout: halves carry K and K+1 vs K+2,K+3
      v2f A0 = *reinterpret_cast<const v2f*>(&wlds[(ln)      * ROWQ + kof]);
      v2f A1 = *reinterpret_cast<const v2f*>(&wlds[(16 + ln) * ROWQ + kof]);
      acc0 = __builtin_amdgcn_wmma_f32_16x16x4_f32(false, A0, false, Bm[t],
                                                   (short)0, acc0, false, false);
      acc1 = __builtin_amdgcn_wmma_f32_16x16x4_f32(false, A1, false, Bm[t],
                                                   (short)0, acc1, false, false);
    }

    // ---- reverse marshal: D (M=sample rows, N=feat lanes) -> sample-per-lane ----
    {
      float* st0 = &wlds[((low ? 0 : 8)) * ROWQ + ln];        // samples 0..15
      float* st1 = st0 + 16 * ROWQ;                           // samples 16..31
      #pragma unroll
      for (int r = 0; r < 8; ++r) {
        st0[r * ROWQ] = acc0[r];
        st1[r * ROWQ] = acc1[r];
      }
    }
    float4 lo = rowp[0];   // Re y0..3 of this lane's sample
    float4 hi = rowp[1];   // Im y0..3

    // Pade 2 on 4 complex units
    cplx y0 = pade({lo.x, hi.x}, a2, q2);
    cplx y1 = pade({lo.y, hi.y}, a2, q2);
    cplx y2 = pade({lo.z, hi.z}, a2, q2);
    cplx y3 = pade({lo.w, hi.w}, a2, q2);

    // layer 3 (4->1 complex)
    cplx o = bb3;
    o = cfma(y0, w3[0], o);
    o = cfma(y1, w3[1], o);
    o = cfma(y2, w3[2], o);
    o = cfma(y3, w3[3], o);

    reinterpret_cast<float2*>(out)[s] = make_float2(o.re, o.im);
  }
}

extern "C" void kernel_launch(void* const* d_in, const int* in_sizes, int n_in,
                              void* d_out, int out_size, void* d_ws, size_t ws_size,
                              hipStream_t stream) {
  const float* x   = (const float*)d_in[0];
  const float* W1r = (const float*)d_in[1];
  const float* W1i = (const float*)d_in[2];
  const float* b1r = (const float*)d_in[3];
  const float* b1i = (const float*)d_in[4];
  const float* W2r = (const float*)d_in[5];
  const float* W2i = (const float*)d_in[6];
  const float* b2r = (const float*)d_in[7];
  const float* b2i = (const float*)d_in[8];
  const float* W3r = (const float*)d_in[9];
  const float* W3i = (const float*)d_in[10];
  const float* b3r = (const float*)d_in[11];
  const float* b3i = (const float*)d_in[12];
  const float* a1r = (const float*)d_in[13];
  const float* a1i = (const float*)d_in[14];
  const float* q1r = (const float*)d_in[15];
  const float* q1i = (const float*)d_in[16];
  const float* a2r = (const float*)d_in[17];
  const float* a2i = (const float*)d_in[18];
  const float* q2r = (const float*)d_in[19];
  const float* q2i = (const float*)d_in[20];
  float* out = (float*)d_out;

  int n = in_sizes[0] / 2;                 // samples
  int blocks = (n + 255) / 256;
  if (blocks > 2048) blocks = 2048;        // persistent-ish: amortize weight setup
  if (blocks < 1) blocks = 1;

  pade_net_kernel<<<dim3(blocks), dim3(256), 0, stream>>>(
      x, W1r, W1i, b1r, b1i, W2r, W2i, b2r, b2i, W3r, W3i, b3r, b3i,
      a1r, a1i, q1r, q1i, a2r, a2i, q2r, q2i, out, n);
}